// MultiHeadAttention_21449066676608
// MI455X (gfx1250) — compile-verified
//
#include <hip/hip_runtime.h>
#include <hip/hip_bf16.h>
#include <cmath>

#define BB 32
#define TT 512
#define DD 512
#define HH 8
#define DHH 64
#define NEGV (-4294967295.0f)

typedef __attribute__((ext_vector_type(2))) float v2f;
typedef __attribute__((ext_vector_type(8))) float v8f;

// D = A(16x4,f32) * B(4x16,f32) + C(16x16,f32)
static __device__ __forceinline__ v8f wmma4(v2f a, v2f b, v8f c) {
  return __builtin_amdgcn_wmma_f32_16x16x4_f32(false, a, false, b, (short)0, c,
                                               false, false);
}

// ---------------------------------------------------------------------------
// Kernel 1: key/query padding masks: mask = (rowsum != 0) ? 1 : 0
// one wave (32 lanes) per row of 512 floats
// ---------------------------------------------------------------------------
__global__ void mask_kernel(const float* __restrict__ q,
                            const float* __restrict__ k,
                            float* __restrict__ qmask,
                            float* __restrict__ kmask) {
  const int gwave = (int)((blockIdx.x * blockDim.x + threadIdx.x) >> 5);
  const int lane = threadIdx.x & 31;
  if (gwave >= BB * TT) return;
  const float4* qr = (const float4*)(q + (size_t)gwave * DD);
  const float4* kr = (const float4*)(k + (size_t)gwave * DD);
  float sq = 0.f, sk = 0.f;
  for (int i = lane; i < DD / 4; i += 32) {
    float4 a = qr[i];
    float4 b = kr[i];
    sq += (a.x + a.y) + (a.z + a.w);
    sk += (b.x + b.y) + (b.z + b.w);
  }
  for (int off = 16; off > 0; off >>= 1) {
    sq += __shfl_xor(sq, off, 32);
    sk += __shfl_xor(sk, off, 32);
  }
  if (lane == 0) {
    qmask[gwave] = (sq != 0.f) ? 1.f : 0.f;
    kmask[gwave] = (sk != 0.f) ? 1.f : 0.f;
  }
}

// ---------------------------------------------------------------------------
// Kernel 2: QKV projection.  One workgroup per (batch, 16-row tile).
// LDS holds the 16x512 query panel and 16x512 key panel.  Wave w owns
// output columns [64w, 64w+64) of each of Q, K, V: per matrix it computes
// 4 adjacent 16x16 tiles simultaneously so each A-fragment LDS load feeds
// 4 v_wmma_f32_16x16x4_f32 ops.
// torch Linear: out[t,o] = sum_i x[t,i] * W[o,i]  (W row-major (out,in))
// ---------------------------------------------------------------------------
__global__ void __launch_bounds__(256) qkv_kernel(
    const float* __restrict__ qin, const float* __restrict__ kin,
    const float* __restrict__ Wq, const float* __restrict__ bq,
    const float* __restrict__ Wk, const float* __restrict__ bk,
    const float* __restrict__ Wv, const float* __restrict__ bv,
    float* __restrict__ Qws, float* __restrict__ Kws, float* __restrict__ Vws) {
  __shared__ float ldsQ[16 * DD];
  __shared__ float ldsK[16 * DD];
  const int bb = blockIdx.x / (TT / 16);
  const int row0 = (blockIdx.x % (TT / 16)) * 16;
  const int tid = threadIdx.x;

  const float4* qsrc = (const float4*)(qin + (size_t)(bb * TT + row0) * DD);
  const float4* ksrc = (const float4*)(kin + (size_t)(bb * TT + row0) * DD);
  for (int i = tid; i < 16 * DD / 4; i += 256) {
    ((float4*)ldsQ)[i] = qsrc[i];
    ((float4*)ldsK)[i] = ksrc[i];
  }
  __syncthreads();

  const int wave = tid >> 5;
  const int lane = tid & 31;
  const int n = lane & 15;         // A-frag row m / B-frag col n / D col n
  const int kb = (lane >> 4) * 2;  // K sub-offset within the 4-chunk
  const int col0 = wave * 64;      // this wave's first output column

  for (int mat = 0; mat < 3; ++mat) {
    const float* A = (mat == 0) ? ldsQ : ldsK;
    const float* W = (mat == 0) ? Wq : (mat == 1) ? Wk : Wv;
    const float* bias = (mat == 0) ? bq : (mat == 1) ? bk : bv;
    float* Out = (mat == 0) ? Qws : (mat == 1) ? Kws : Vws;

    const float* arow = A + n * DD;  // X row m (m == lane&15)
    const float* wrow0 = W + (size_t)(col0 + 0 * 16 + n) * DD;
    const float* wrow1 = W + (size_t)(col0 + 1 * 16 + n) * DD;
    const float* wrow2 = W + (size_t)(col0 + 2 * 16 + n) * DD;
    const float* wrow3 = W + (size_t)(col0 + 3 * 16 + n) * DD;
    __builtin_prefetch(wrow0, 0, 0);
    __builtin_prefetch(wrow1, 0, 0);
    __builtin_prefetch(wrow2, 0, 0);
    __builtin_prefetch(wrow3, 0, 0);

    v8f c0 = {}, c1 = {}, c2 = {}, c3 = {};
#pragma unroll 2
    for (int k0 = 0; k0 < DD; k0 += 4) {
      const v2f a = *(const v2f*)(arow + k0 + kb);
      const v2f b0 = *(const v2f*)(wrow0 + k0 + kb);
      const v2f b1 = *(const v2f*)(wrow1 + k0 + kb);
      const v2f b2 = *(const v2f*)(wrow2 + k0 + kb);
      const v2f b3 = *(const v2f*)(wrow3 + k0 + kb);
      c0 = wmma4(a, b0, c0);
      c1 = wmma4(a, b1, c1);
      c2 = wmma4(a, b2, c2);
      c3 = wmma4(a, b3, c3);
    }

    float* obase = Out + (size_t)(bb * TT + row0) * DD + col0 + n;
#pragma unroll
    for (int t = 0; t < 4; ++t) {
      const v8f c = (t == 0) ? c0 : (t == 1) ? c1 : (t == 2) ? c2 : c3;
      const float bia = bias[col0 + t * 16 + n];
      float* orow = obase + t * 16;
#pragma unroll
      for (int j = 0; j < 8; ++j) {
        const int m = j + 8 * (lane >> 4);
        orow[(size_t)m * DD] = c[j] + bia;
      }
    }
  }
}

// ---------------------------------------------------------------------------
// Kernel 3: causal flash attention + residual + LayerNorm.
// One workgroup per (batch, 16-query tile); wave w = head w (identical
// causal trip counts across waves -> __syncthreads is safe).
// ---------------------------------------------------------------------------
__global__ void __launch_bounds__(256) attn_ln_kernel(
    const float* __restrict__ Qws, const float* __restrict__ Kws,
    const float* __restrict__ Vws, const float* __restrict__ qin,
    const float* __restrict__ qmask, const float* __restrict__ kmask,
    const float* __restrict__ gamma, const float* __restrict__ beta,
    float* __restrict__ out) {
  __shared__ float ldsP[8][16 * 16];  // per-wave P transpose scratch
  __shared__ float ldsO[16 * DD];     // per-tile attention output (all heads)

  const int bb = blockIdx.x / (TT / 16);
  const int qt = blockIdx.x % (TT / 16);
  const int qrow0 = qt * 16;
  const int wave = threadIdx.x >> 5;
  const int lane = threadIdx.x & 31;
  const int head = wave;
  const int n = lane & 15;
  const int hi = lane >> 4;
  const int kb = hi * 2;

  // Preload Q A-fragments (16x64), pre-scaled by 1/sqrt(DH) = 1/8
  v2f aQ[16];
  {
    const float* qbase =
        Qws + (size_t)(bb * TT + qrow0 + n) * DD + head * DHH + kb;
#pragma unroll
    for (int kc = 0; kc < 16; ++kc) {
      v2f a = *(const v2f*)(qbase + kc * 4);
      aQ[kc] = a * 0.125f;
    }
  }

  float m_run[8], l_run[8];
  v8f o[4];
  const v8f vzero = {};
#pragma unroll
  for (int j = 0; j < 8; ++j) {
    m_run[j] = -INFINITY;
    l_run[j] = 0.f;
  }
#pragma unroll
  for (int od = 0; od < 4; ++od) o[od] = vzero;

  for (int kt = 0; kt <= qt; ++kt) {
    const int key0 = kt * 16;

    // S = (Q/8) @ K^T : 16 WMMA K-steps over DH=64
    v8f s = vzero;
    const float* kbase =
        Kws + (size_t)(bb * TT + key0 + n) * DD + head * DHH + kb;
#pragma unroll
    for (int kc = 0; kc < 16; ++kc) {
      v2f b = *(const v2f*)(kbase + kc * 4);
      s = wmma4(aQ[kc], b, s);
    }

    // key-padding + causal masks, then online softmax update
    const float km = kmask[bb * TT + key0 + n];
    float sv[8], rmax[8], p[8], alpha[8];
#pragma unroll
    for (int j = 0; j < 8; ++j) {
      const int m = j + 8 * hi;
      const bool ok = (km != 0.f) && (key0 + n <= qrow0 + m);
      sv[j] = ok ? s[j] : NEGV;
      float v = sv[j];
#pragma unroll
      for (int off = 8; off > 0; off >>= 1) v = fmaxf(v, __shfl_xor(v, off, 16));
      rmax[j] = v;
    }
#pragma unroll
    for (int j = 0; j < 8; ++j) {
      const float mnew = fmaxf(m_run[j], rmax[j]);
      alpha[j] = __expf(m_run[j] - mnew);
      p[j] = __expf(sv[j] - mnew);
      m_run[j] = mnew;
      float rs = p[j];
#pragma unroll
      for (int off = 8; off > 0; off >>= 1) rs += __shfl_xor(rs, off, 16);
      l_run[j] = l_run[j] * alpha[j] + rs;
    }
#pragma unroll
    for (int od = 0; od < 4; ++od)
#pragma unroll
      for (int j = 0; j < 8; ++j) o[od][j] *= alpha[j];

    // Transpose P (C/D layout) -> A layout via per-wave LDS scratch.
    // LDS ops are in-order within a wave; the volatile asm stops the
    // compiler reordering and drains DScnt.
#pragma unroll
    for (int j = 0; j < 8; ++j) ldsP[wave][(j + 8 * hi) * 16 + n] = p[j];
    asm volatile("s_wait_dscnt 0" ::: "memory");

    v2f aP[4];
    const float* prow = &ldsP[wave][n * 16 + kb];
#pragma unroll
    for (int kc = 0; kc < 4; ++kc) aP[kc] = *(const v2f*)(prow + kc * 4);

    // O += P @ V  (4 WMMA K-steps x 4 output dh-tiles)
    const float* vbase = Vws + (size_t)(bb * TT + key0) * DD + head * DHH;
#pragma unroll
    for (int kc = 0; kc < 4; ++kc) {
      const float* vr0 = vbase + (size_t)(kc * 4 + kb) * DD;
#pragma unroll
      for (int od = 0; od < 4; ++od) {
        v2f b;
        b.x = vr0[od * 16 + n];
        b.y = vr0[DD + od * 16 + n];
        o[od] = wmma4(aP[kc], b, o[od]);
      }
    }
  }

  // softmax denominator + query padding mask, stash rows in LDS
  float fac[8];
#pragma unroll
  for (int j = 0; j < 8; ++j) {
    const int m = j + 8 * hi;
    fac[j] = qmask[bb * TT + qrow0 + m] / l_run[j];
  }
#pragma unroll
  for (int od = 0; od < 4; ++od)
#pragma unroll
    for (int j = 0; j < 8; ++j)
      ldsO[(j + 8 * hi) * DD + head * DHH + od * 16 + n] = o[od][j] * fac[j];
  __syncthreads();

  // residual + LayerNorm (unbiased std, eps added to std); wave w -> rows 2w,2w+1
  for (int rr = 0; rr < 2; ++rr) {
    const int r = wave * 2 + rr;
    const float* xrow = ldsO + r * DD;
    const float* qres = qin + (size_t)(bb * TT + qrow0 + r) * DD;
    float xv[16];
    float sum = 0.f, sumsq = 0.f;
#pragma unroll
    for (int i = 0; i < 16; ++i) {
      const int idx = lane + i * 32;
      const float x = xrow[idx] + qres[idx];
      xv[i] = x;
      sum += x;
      sumsq += x * x;
    }
#pragma unroll
    for (int off = 16; off > 0; off >>= 1) {
      sum += __shfl_xor(sum, off, 32);
      sumsq += __shfl_xor(sumsq, off, 32);
    }
    const float mean = sum * (1.0f / DD);
    const float var =
        fmaxf(0.f, (sumsq - (float)DD * mean * mean) * (1.0f / (DD - 1)));
    const float inv = 1.0f / (sqrtf(var) + 1e-8f);
    float* orow = out + (size_t)(bb * TT + qrow0 + r) * DD;
#pragma unroll
    for (int i = 0; i < 16; ++i) {
      const int idx = lane + i * 32;
      orow[idx] = gamma[idx] * (xv[i] - mean) * inv + beta[idx];
    }
  }
}

// ---------------------------------------------------------------------------
extern "C" void kernel_launch(void* const* d_in, const int* in_sizes, int n_in,
                              void* d_out, int out_size, void* d_ws,
                              size_t ws_size, hipStream_t stream) {
  const float* queries = (const float*)d_in[0];
  const float* keys = (const float*)d_in[1];
  const float* Wq = (const float*)d_in[2];
  const float* bq = (const float*)d_in[3];
  const float* Wk = (const float*)d_in[4];
  const float* bk = (const float*)d_in[5];
  const float* Wv = (const float*)d_in[6];
  const float* bv = (const float*)d_in[7];
  const float* gamma = (const float*)d_in[8];
  const float* beta = (const float*)d_in[9];

  float* ws = (float*)d_ws;
  const size_t btd = (size_t)BB * TT * DD;
  float* Qws = ws;
  float* Kws = ws + btd;
  float* Vws = ws + 2 * btd;
  float* qmask = ws + 3 * btd;
  float* kmask = qmask + (size_t)BB * TT;

  mask_kernel<<<(BB * TT * 32 + 255) / 256, 256, 0, stream>>>(queries, keys,
                                                              qmask, kmask);
  qkv_kernel<<<BB * (TT / 16), 256, 0, stream>>>(queries, keys, Wq, bq, Wk, bk,
                                                 Wv, bv, Qws, Kws, Vws);
  attn_ln_kernel<<<BB * (TT / 16), 256, 0, stream>>>(
      Qws, Kws, Vws, queries, qmask, kmask, gamma, beta, (float*)d_out);
}